// VoxelRepresentation_9715216023992
// MI455X (gfx1250) — compile-verified
//
#include <hip/hip_runtime.h>

typedef __attribute__((ext_vector_type(2))) float v2f;
typedef __attribute__((ext_vector_type(8))) float v8f;

#define GSZ        32
#define G3         32768
#define NNB        16
#define ATOM_TYPE  6
#define LOG2E      1.44269504088896340736f

// Tile = 16 consecutive voxels along z (z is the fastest-varying output dim).
// WMMA: M = neighbor (16), N = voxel-in-tile (16), K = 4 (x,y,z,const).
//   A[n][k]   = (-2c*dx, -2c*dy, -2c*dz, mask ? c*||d||^2 : -1e30)
//   B[k][v]   = (gx, gy, gz, 1)
//   C[n][v]   = c*||g_v||^2   (row-independent)
//   D[n][v]   = c*||g_v - d_n||^2   (in exp2 domain; c includes log2 e)
// D layout (32-bit 16x16): VGPR r, lanes 0-15 -> (M=r,   N=lane),
//                                   lanes16-31 -> (M=r+8, N=lane-16).
// => lane v holds neighbors 0..7 of voxel v, lane v+16 holds neighbors 8..15.
//    out[v] = in-lane sum of 8 exp2's  +  SWAPX16 partner's sum.
__global__ __launch_bounds__(256) void voxel_wmma_kernel(
    const float* __restrict__ dv,     // (4,32,16,3) f32
    const int*   __restrict__ an,     // (32,16) i32
    const float* __restrict__ sg,     // (1) f32
    float*       __restrict__ out)    // (4,32,32,32,32) f32
{
    const int tid   = threadIdx.x;
    const int lane  = tid & 31;
    const int wave  = tid >> 5;
    const int slab  = blockIdx.x >> 3;   // b*32 + a   (0..127)
    const int chunk = blockIdx.x & 7;    // 4096-voxel chunk within the slab
    const int a     = slab & 31;

    const float s    = sg[0];
    const float c2   = (-0.5f / (s * s)) * LOG2E;  // exp2-domain coefficient
    const float step = 8.0f / 31.0f;               // linspace(-4, 4, 32)

    __shared__ float4 s_bv[NNB];

    // Stage the per-(b,a) neighbor matrix rows once (mask folded in).
    if (tid < NNB) {
        const int n = tid;
        const float dx = dv[(slab * NNB + n) * 3 + 0];
        const float dy = dv[(slab * NNB + n) * 3 + 1];
        const float dz = dv[(slab * NNB + n) * 3 + 2];
        const bool  m  = (an[a * NNB + n] == ATOM_TYPE);
        float4 q;
        q.x = -2.0f * c2 * dx;
        q.y = -2.0f * c2 * dy;
        q.z = -2.0f * c2 * dz;
        q.w = m ? c2 * (dx * dx + dy * dy + dz * dz) : -1.0e30f; // exp2 -> 0
        s_bv[n] = q;
    }
    __syncthreads();

    // A operand (16x4 f32): lanes 0-15 hold K={0,1}, lanes 16-31 hold K={2,3}.
    const float4 q = s_bv[lane & 15];
    v2f amat;
    amat.x = (lane < 16) ? q.x : q.z;
    amat.y = (lane < 16) ? q.y : q.w;

    const int col = lane & 15;  // this lane's voxel column within a tile

    // 8 waves x 32 tiles x 16 voxels = 4096 voxels per block.
    for (int i = 0; i < 32; ++i) {
        const int tile    = wave * 32 + i;
        const int voxBase = chunk * 4096 + tile * 16;   // within slab
        const int x  =  voxBase >> 10;
        const int y  = (voxBase >> 5) & 31;
        const int z0 =  voxBase & 31;                   // 0 or 16

        const float gx = -4.0f + step * (float)x;
        const float gy = -4.0f + step * (float)y;
        const float gz = -4.0f + step * (float)(z0 + col);

        // B operand (4x16 f32), column = voxel.
        v2f bmat;
        bmat.x = (lane < 16) ? gx : gz;
        bmat.y = (lane < 16) ? gy : 1.0f;

        // C = c*||g_v||^2 broadcast down the 8 rows this lane holds.
        const float gn = c2 * (gx * gx + gy * gy + gz * gz);
        v8f cacc = {gn, gn, gn, gn, gn, gn, gn, gn};

        v8f dmat = __builtin_amdgcn_wmma_f32_16x16x4_f32(
            false, amat, false, bmat, (short)0, cacc, false, false);

        // exp2 over the 8 neighbor scores held in-lane, pairwise-tree sum.
        float e0 = __builtin_amdgcn_exp2f(dmat[0]);
        float e1 = __builtin_amdgcn_exp2f(dmat[1]);
        float e2 = __builtin_amdgcn_exp2f(dmat[2]);
        float e3 = __builtin_amdgcn_exp2f(dmat[3]);
        float e4 = __builtin_amdgcn_exp2f(dmat[4]);
        float e5 = __builtin_amdgcn_exp2f(dmat[5]);
        float e6 = __builtin_amdgcn_exp2f(dmat[6]);
        float e7 = __builtin_amdgcn_exp2f(dmat[7]);
        float acc = ((e0 + e1) + (e2 + e3)) + ((e4 + e5) + (e6 + e7));

        // Combine half-wave partials: lane v (+) lane v+16  (SWAPX16 swizzle:
        // group-of-32, and=0x1f, or=0, xor=0x10 -> offset 0x401F).
        acc += __int_as_float(
            __builtin_amdgcn_ds_swizzle(__float_as_int(acc), 0x401F));

        // Lanes 0-15 hold the 16 finished voxels: one coalesced 64B NT store.
        if (lane < 16) {
            __builtin_nontemporal_store(acc, &out[slab * G3 + voxBase + col]);
        }
    }
}

extern "C" void kernel_launch(void* const* d_in, const int* in_sizes, int n_in,
                              void* d_out, int out_size, void* d_ws, size_t ws_size,
                              hipStream_t stream) {
    (void)in_sizes; (void)n_in; (void)d_ws; (void)ws_size; (void)out_size;
    const float* dv = (const float*)d_in[0];   // (4,32,16,3)
    const int*   an = (const int*)d_in[1];     // (32,16)
    const float* sg = (const float*)d_in[2];   // (1,)
    float*       o  = (float*)d_out;           // (4,32,32,32,32)
    // 128 slabs x 8 chunks = 1024 blocks, 256 threads (8 wave32) each.
    voxel_wmma_kernel<<<dim3(1024), dim3(256), 0, stream>>>(dv, an, sg, o);
}